// XD_18330920419781
// MI455X (gfx1250) — compile-verified
//
#include <hip/hip_runtime.h>

typedef __attribute__((ext_vector_type(2))) float v2f;
typedef __attribute__((ext_vector_type(8))) float v8f;

#define H 256
#define W 256
#define CI 16
#define CO 16
#define RPB 8                 // output rows per block
#define TR (RPB + 2)          // 10 tile rows (halo)
#define TC (16 + 2)           // 18 tile cols (halo)
#define CIP 20                // padded channel pitch: conflict-free b64 B loads
#define NITER 36              // 9 taps * 4 ci-slices of K=4

#if __has_builtin(__builtin_amdgcn_global_load_async_to_lds_b32) && \
    __has_builtin(__builtin_amdgcn_s_wait_asynccnt)
#define USE_ASYNC 1
typedef __attribute__((address_space(1))) int gint_t;   // global AS ("__device__ int *")
typedef __attribute__((address_space(3))) int lint_t;   // LDS AS
#else
#define USE_ASYNC 0
#endif

// y[b,o,i,j] = sum_ci sum_{ta,tb} w[o,ci,ta,tb] * x[b,ci,(i+1-ta)&255,(j+1-tb)&255]
// D[16co x 16col] += A[16co x 4ci] * B[4ci x 16col], 36x V_WMMA_F32_16X16X4_F32 / wave.
__global__ __launch_bounds__(256) void xd_circconv3x3_wmma(
    const float* __restrict__ x, const float* __restrict__ wgt, float* __restrict__ y)
{
    __shared__ float wtile[NITER * 64];      // A operands pre-swizzled: [iter][lane][2]
    __shared__ float xtile[TR * TC * CIP];   // haloed x tile, channel innermost

    const int tid  = threadIdx.x;
    const int lane = tid & 31;
    const int wave = tid >> 5;

    const int col0  = blockIdx.x * 16;
    const int row0  = blockIdx.y * RPB;
    const int batch = blockIdx.z;

    // ---- stage weights into WMMA A layout:
    // 16x4 f32 A: lane<16 -> M=lane, K={k0,k0+1}; lane>=16 -> K={k0+2,k0+3}
    for (int e = tid; e < NITER * 64; e += 256) {
        const int p   = e & 1;          // which of the lane's two K values
        const int l   = (e >> 1) & 31;  // lane
        const int it  = e >> 6;         // iter = tap*4 + ks
        const int tap = it >> 2, ks = it & 3;
        const int m   = l & 15;
        const int kb  = (l >> 4) << 1;
        const int ci  = ks * 4 + kb + p;
        wtile[e] = wgt[((m * CI + ci) * 3 + (tap / 3)) * 3 + (tap % 3)];
    }

    // ---- stage haloed x tile (circular wrap = &255), ci innermost with pitch 20
    const float* xb = x + (size_t)batch * CI * H * W;
    for (int e = tid; e < CI * TR * TC; e += 256) {
        const int ci  = e / (TR * TC);
        const int rem = e - ci * (TR * TC);
        const int rr  = rem / TC;
        const int cc  = rem - rr * TC;
        const int gr  = (row0 - 1 + rr) & (H - 1);
        const int gc  = (col0 - 1 + cc) & (W - 1);
        const float* g = xb + (ci * H + gr) * W + gc;
        float* l = &xtile[(rr * TC + cc) * CIP + ci];
#if USE_ASYNC
        __builtin_amdgcn_global_load_async_to_lds_b32(
            (gint_t*)const_cast<float*>(g), (lint_t*)l, 0, 0);
#else
        *l = *g;
#endif
    }
#if USE_ASYNC
    __builtin_amdgcn_s_wait_asynccnt(0);
#endif
    __syncthreads();

    // ---- WMMA accumulation: wave owns output row row0+wave, 16 cols
    const int i  = wave;
    const int n  = lane & 15;          // B/D column
    const int kb = (lane >> 4) << 1;   // K sub-offset per half-wave
    const float* abase = &wtile[lane * 2];

    v8f c0 = {}, c1 = {};              // two chains: no WMMA->WMMA RAW hazard
#pragma unroll
    for (int tap = 0; tap < 9; ++tap) {
        const int ta = tap / 3, tb = tap % 3;
        const int rr = i + 2 - ta;     // LDS row of input (i+1-ta)
        const int cc = n + 2 - tb;     // LDS col of input (j+1-tb)
        const float* bbase = &xtile[(rr * TC + cc) * CIP + kb];
#pragma unroll
        for (int ks = 0; ks < 4; ++ks) {
            const int it = tap * 4 + ks;
            const v2f a = *(const v2f*)(abase + it * 64);   // ds_load_b64, imm offset
            const v2f b = *(const v2f*)(bbase + ks * 4);    // ds_load_b64, imm offset
            if (it & 1)
                c1 = __builtin_amdgcn_wmma_f32_16x16x4_f32(
                        false, a, false, b, (short)0, c1, false, false);
            else
                c0 = __builtin_amdgcn_wmma_f32_16x16x4_f32(
                        false, a, false, b, (short)0, c0, false, false);
        }
    }
    const v8f c = c0 + c1;

    // ---- store D: lane<16 -> M=v, lane>=16 -> M=v+8; N = lane&15
    const int row   = row0 + i;
    const int obase = (lane >> 4) * 8;
    float* yb = y + (size_t)batch * CO * H * W;
#pragma unroll
    for (int v = 0; v < 8; ++v) {
        yb[((obase + v) * H + row) * W + col0 + n] = c[v];
    }
}

extern "C" void kernel_launch(void* const* d_in, const int* in_sizes, int n_in,
                              void* d_out, int out_size, void* d_ws, size_t ws_size,
                              hipStream_t stream) {
    const float* x = (const float*)d_in[0];   // [16,16,256,256] f32
    const float* w = (const float*)d_in[1];   // [16,16,3,3] f32
    float* y = (float*)d_out;                 // [16,16,256,256] f32

    dim3 grid(W / 16, H / RPB, 16);           // 16 x 32 x 16 = 8192 blocks
    dim3 block(256);                           // 8 waves (wave32)
    hipLaunchKernelGGL(xd_circconv3x3_wmma, grid, block, 0, stream, x, w, y);
}